// ParametricEqualizer_4157528342990
// MI455X (gfx1250) — compile-verified
//
#include <hip/hip_runtime.h>
#include <hip/hip_bf16.h>

// WMMA operand vector types (wave32, gfx1250):
//  A/B for V_WMMA_F32_16X16X4_F32: 64 f32 / 32 lanes = 2 VGPRs -> v2f
//  C/D 16x16 f32: 256 f32 / 32 lanes = 8 VGPRs -> v8f
typedef __attribute__((ext_vector_type(2))) float v2f;
typedef __attribute__((ext_vector_type(8))) float v8f;

#define NFREQ     1025      // WINDOWS/2 + 1
#define ROW_TILES 1024      // (64*256)/16
#define COL_TILES 65        // ceil(1025/16)
#define COL_CHUNKS 13       // column-tile parallelism (65 = 13*5)

// 2*pi/SR, 2*pi/WINDOWS, ln(10)/40
#define W0_SCALE   2.6179938779914946e-4f
#define TH_SCALE   3.0679615757712823e-3f
#define GAIN_SCALE 0.05756462732485114f

__global__ __launch_bounds__(256)
void ParametricEqualizer_4157528342990_kernel(const float* __restrict__ center,
                                              const float* __restrict__ gain,
                                              const float* __restrict__ q,
                                              float* __restrict__ out) {
    const int lane  = threadIdx.x & 31;
    const int wave  = threadIdx.x >> 5;
    const int rowTile = blockIdx.x * 8 + wave;   // 0..1023
    const int laneN = lane & 15;
    const bool hi   = lane >= 16;

    // ---- Per-row biquad coefficients (RBJ peaking EQ). Each lane computes
    // the coefficients for row (lane & 15); upper lanes duplicate lower
    // lanes' rows so the A-matrix lane-halves can be filled by selects.
    const int row = rowTile * 16 + laneN;        // < 16384, always in range
    const float fc = center[row];
    const float g  = gain[row];
    const float Q  = q[row];

    const float w0 = W0_SCALE * fc;
    float sw0, cw0;
    sincosf(w0, &sw0, &cw0);
    const float alpha = sw0 / (2.0f * Q);
    const float Aamp  = expf(g * GAIN_SCALE);
    const float aDivA = alpha / Aamp;
    const float aMulA = alpha * Aamp;
    const float a0 = 1.0f + aDivA;
    const float a1 = -2.0f * cw0;
    const float a2 = 1.0f - aDivA;
    const float b0 = 1.0f + aMulA;
    const float b1 = a1;
    const float b2 = 1.0f - aMulA;

    // A-matrix (16x4, K=3 padded to 4 with zero):
    //   lanes 0-15:  {K0,K1} = {c0, c1};  lanes 16-31: {K2,K3} = {c2, 0}
    v2f Afir, Aiir;
    Afir[0] = hi ? b2 : b0;   Afir[1] = hi ? 0.0f : b1;
    Aiir[0] = hi ? a2 : a0;   Aiir[1] = hi ? 0.0f : a1;

    const int rbase = rowTile * 16 + (hi ? 8 : 0);   // D-matrix row base for this lane half
    float2* __restrict__ outc = reinterpret_cast<float2*>(out);

    for (int jt = blockIdx.y; jt < COL_TILES; jt += COL_CHUNKS) {
        const int k = jt * 16 + laneN;               // this lane's frequency column
        const float theta = TH_SCALE * (float)k;     // z = exp(-i*theta)
        float s1, c1;
        sincosf(theta, &s1, &c1);
        const float c2 = c1 * c1 - s1 * s1;          // cos(2*theta)
        const float s2 = 2.0f * s1 * c1;             // sin(2*theta)

        // B-matrix (4x16): lanes 0-15 hold rows {K0,K1} for column lane,
        // lanes 16-31 hold rows {K2,K3}.  z^0=1, z^1=c1-i*s1, z^2=c2-i*s2.
        v2f Bre, Bim;
        Bre[0] = hi ?  c2  : 1.0f;   Bre[1] = hi ? 0.0f :  c1;
        Bim[0] = hi ? -s2  : 0.0f;   Bim[1] = hi ? 0.0f : -s1;

        const v8f zero = {};
        // 8 args: (neg_a, A, neg_b, B, c_mod, C, reuse_a, reuse_b)
        v8f fr = __builtin_amdgcn_wmma_f32_16x16x4_f32(false, Afir, false, Bre, (short)0, zero, false, false);
        v8f fi = __builtin_amdgcn_wmma_f32_16x16x4_f32(false, Afir, false, Bim, (short)0, zero, false, false);
        v8f gr = __builtin_amdgcn_wmma_f32_16x16x4_f32(false, Aiir, false, Bre, (short)0, zero, false, false);
        v8f gi = __builtin_amdgcn_wmma_f32_16x16x4_f32(false, Aiir, false, Bim, (short)0, zero, false, false);

        // Complex division + store; only the store is predicated so EXEC is
        // all-1s at the WMMAs above.
        if (k < NFREQ) {
            #pragma unroll
            for (int r = 0; r < 8; ++r) {
                const float dr = gr[r], di = gi[r];
                const float inv = 1.0f / (dr * dr + di * di);
                const float nr = fr[r], ni = fi[r];
                float2 h;
                h.x = (nr * dr + ni * di) * inv;
                h.y = (ni * dr - nr * di) * inv;
                outc[(size_t)(rbase + r) * NFREQ + (size_t)k] = h;
            }
        }
    }
}

extern "C" void kernel_launch(void* const* d_in, const int* in_sizes, int n_in,
                              void* d_out, int out_size, void* d_ws, size_t ws_size,
                              hipStream_t stream) {
    const float* center = (const float*)d_in[0];
    const float* gain   = (const float*)d_in[1];
    const float* q      = (const float*)d_in[2];
    float* out          = (float*)d_out;   // complex64 -> interleaved (re, im) floats

    // 8 waves per block, one row-tile per wave; 13-way column parallelism.
    dim3 grid(ROW_TILES / 8, COL_CHUNKS, 1);
    dim3 block(256, 1, 1);
    ParametricEqualizer_4157528342990_kernel<<<grid, block, 0, stream>>>(center, gain, q, out);
}